// APPNP_55808805044378
// MI455X (gfx1250) — compile-verified
//
#include <hip/hip_runtime.h>
#include <math.h>

// ---------------- problem constants (match reference) ----------------
#define N_NODES   100000
#define N_EDGES   3200000
#define F_IN      512
#define NHID      64
#define NCLS      40
#define NCLS_PAD  48
#define ALPHA     0.1f
#define KPROP     10

#define ROW_TILES (N_NODES / 16)       // 6250, exact
#define XS_STRIDE 68                   // 64 cols padded -> conflict-free ds_load_b64

typedef __attribute__((ext_vector_type(2))) float v2f;
typedef __attribute__((ext_vector_type(8))) float v8f;

// =====================================================================
// Fused MLP: h[N,40] = (relu(feat @ W1 + b1)) @ W2 + b2
// One wave per 16-row tile. GEMM1 uses V_WMMA_F32_16X16X4_F32 over K=512
// (4 col tiles of NHID=64). Result is transposed through LDS (per-wave
// private region, DS ops are in-order within a wave -> no barrier) and fed
// straight into GEMM2 (K=64, 3 col tiles covering NCLS padded to 48).
// =====================================================================
__global__ __launch_bounds__(256, 2) void appnp_mlp_kernel(
    const float* __restrict__ feat, const float* __restrict__ W1,
    const float* __restrict__ b1,   const float* __restrict__ W2,
    const float* __restrict__ b2,   float* __restrict__ h)
{
    __shared__ float xs[8][16 * XS_STRIDE];   // per-wave X1 tile (relu output)
    __shared__ float w2s[64 * NCLS_PAD];      // W2 zero-padded to 64x48
    __shared__ float b2s[NCLS_PAD];

    const int tid = threadIdx.x;

    // stage padded W2 / b2 once per block
    for (int i = tid; i < 64 * NCLS_PAD; i += 256) {
        int r = i / NCLS_PAD, c = i % NCLS_PAD;
        w2s[i] = (c < NCLS) ? W2[r * NCLS + c] : 0.0f;
    }
    if (tid < NCLS_PAD) b2s[tid] = (tid < NCLS) ? b2[tid] : 0.0f;
    __syncthreads();

    const int wave    = tid >> 5;
    const int lane    = tid & 31;
    const int rowTile = blockIdx.x * 8 + wave;
    if (rowTile >= ROW_TILES) return;          // wave-uniform, EXEC stays full

    const int lo = lane & 15;                  // 0..15
    const int hi = lane >> 4;                  // 0 or 1

    // ---------------- GEMM1: acc[4] = 16x64 tile of feat @ W1 ----------
    v8f acc[4] = {};
    const float* arow = feat + (size_t)(rowTile * 16 + lo) * F_IN + 2 * hi;

    for (int k0 = 0; k0 < F_IN; k0 += 4) {
        // A fragment: lane holds (row=lo, k = k0+2*hi .. +1)
        v2f a = *(const v2f*)(arow + k0);
        const int bk = k0 + 2 * hi;
        #pragma unroll
        for (int t = 0; t < 4; ++t) {
            v2f b;                              // B fragment of W1 col-tile t
            b.x = W1[(bk    ) * NHID + t * 16 + lo];
            b.y = W1[(bk + 1) * NHID + t * 16 + lo];
            acc[t] = __builtin_amdgcn_wmma_f32_16x16x4_f32(
                false, a, false, b, (short)0, acc[t], false, false);
        }
    }

    // bias + relu -> LDS (row-major 16 x 64, stride 68)
    float bias1[4];
    #pragma unroll
    for (int t = 0; t < 4; ++t) bias1[t] = b1[t * 16 + lo];

    float* xrow = &xs[wave][0];
    #pragma unroll
    for (int t = 0; t < 4; ++t) {
        #pragma unroll
        for (int r = 0; r < 8; ++r) {
            const int m = r + 8 * hi;
            const int n = t * 16 + lo;
            xrow[m * XS_STRIDE + n] = fmaxf(acc[t][r] + bias1[t], 0.0f);
        }
    }
    // same-wave LDS RAW: DS ops are in-order within a wave, no barrier needed

    // ---------------- GEMM2: acc2[3] = 16x48 tile of X1 @ W2 -----------
    v8f acc2[3] = {};
    for (int k0 = 0; k0 < NHID; k0 += 4) {
        v2f a = *(const v2f*)(&xrow[lo * XS_STRIDE + k0 + 2 * hi]);
        const int bk = k0 + 2 * hi;
        #pragma unroll
        for (int t = 0; t < 3; ++t) {
            v2f b;
            b.x = w2s[(bk    ) * NCLS_PAD + t * 16 + lo];
            b.y = w2s[(bk + 1) * NCLS_PAD + t * 16 + lo];
            acc2[t] = __builtin_amdgcn_wmma_f32_16x16x4_f32(
                false, a, false, b, (short)0, acc2[t], false, false);
        }
    }

    // store h (mask padded columns n >= 40)
    #pragma unroll
    for (int t = 0; t < 3; ++t) {
        #pragma unroll
        for (int r = 0; r < 8; ++r) {
            const int m = r + 8 * hi;
            const int n = t * 16 + lo;
            if (n < NCLS)
                h[(size_t)(rowTile * 16 + m) * NCLS + n] = acc2[t][r] + b2s[n];
        }
    }
}

// =====================================================================
// dst = ALPHA * h   (teleport term), float4-vectorized stream
// =====================================================================
__global__ void appnp_init_kernel(float* __restrict__ dst,
                                  const float* __restrict__ h)
{
    const size_t i = (size_t)blockIdx.x * blockDim.x + threadIdx.x;
    const size_t n4 = (size_t)N_NODES * NCLS / 4;
    if (i < n4) {
        float4 v = ((const float4*)h)[i];
        v.x *= ALPHA; v.y *= ALPHA; v.z *= ALPHA; v.w *= ALPHA;
        ((float4*)dst)[i] = v;
    }
}

// =====================================================================
// Edge scatter: dst[row] += (1-ALPHA) * ev * x[col]   (x is L2-resident,
// 16 MB << 192 MB L2; fp32 hardware atomics via unsafeAtomicAdd)
// =====================================================================
__global__ void appnp_edge_kernel(const int* __restrict__ rowp,
                                  const int* __restrict__ colp,
                                  const float* __restrict__ ev,
                                  const float* __restrict__ xin,
                                  float* __restrict__ xout)
{
    const int e = blockIdx.x * blockDim.x + threadIdx.x;
    if (e >= N_EDGES) return;
    const int r = rowp[e];
    const int c = colp[e];
    const float w = (1.0f - ALPHA) * ev[e];
    const float4* src = (const float4*)(xin + (size_t)c * NCLS);
    float* dst = xout + (size_t)r * NCLS;
    #pragma unroll
    for (int q = 0; q < NCLS / 4; ++q) {
        float4 v = src[q];
        unsafeAtomicAdd(dst + q * 4 + 0, w * v.x);
        unsafeAtomicAdd(dst + q * 4 + 1, w * v.y);
        unsafeAtomicAdd(dst + q * 4 + 2, w * v.z);
        unsafeAtomicAdd(dst + q * 4 + 3, w * v.w);
    }
}

// =====================================================================
// log-softmax over the 40 classes of each row (kept in registers)
// =====================================================================
__global__ void appnp_logsoftmax_kernel(const float* __restrict__ x,
                                        float* __restrict__ out)
{
    const int r = blockIdx.x * blockDim.x + threadIdx.x;
    if (r >= N_NODES) return;
    const float4* src = (const float4*)(x + (size_t)r * NCLS);
    float4 v[NCLS / 4];
    float m = -INFINITY;
    #pragma unroll
    for (int q = 0; q < NCLS / 4; ++q) {
        v[q] = src[q];
        m = fmaxf(m, fmaxf(fmaxf(v[q].x, v[q].y), fmaxf(v[q].z, v[q].w)));
    }
    float s = 0.0f;
    #pragma unroll
    for (int q = 0; q < NCLS / 4; ++q) {
        s += __expf(v[q].x - m) + __expf(v[q].y - m)
           + __expf(v[q].z - m) + __expf(v[q].w - m);
    }
    const float ls = m + __logf(s);
    float4* dst = (float4*)(out + (size_t)r * NCLS);
    #pragma unroll
    for (int q = 0; q < NCLS / 4; ++q) {
        float4 o;
        o.x = v[q].x - ls; o.y = v[q].y - ls;
        o.z = v[q].z - ls; o.w = v[q].w - ls;
        dst[q] = o;
    }
}

// =====================================================================
extern "C" void kernel_launch(void* const* d_in, const int* in_sizes, int n_in,
                              void* d_out, int out_size, void* d_ws, size_t ws_size,
                              hipStream_t stream)
{
    const float* feat = (const float*)d_in[0];      // [N, 512]
    const int*   eidx = (const int*)  d_in[1];      // [2, E]
    const float* ev   = (const float*)d_in[2];      // [E]
    const float* W1   = (const float*)d_in[3];      // [512, 64]
    const float* b1   = (const float*)d_in[4];      // [64]
    const float* W2   = (const float*)d_in[5];      // [64, 40]
    const float* b2   = (const float*)d_in[6];      // [40]
    float* out = (float*)d_out;

    float* ws = (float*)d_ws;
    float* h  = ws;                                   // [N,40]
    float* xA = ws + (size_t)N_NODES * NCLS;          // [N,40]
    float* xB = xA + (size_t)N_NODES * NCLS;          // [N,40]

    const int* rowp = eidx;
    const int* colp = eidx + N_EDGES;

    // fused MLP -> h
    appnp_mlp_kernel<<<(ROW_TILES + 7) / 8, 256, 0, stream>>>(
        feat, W1, b1, W2, b2, h);

    // K rounds of personalized propagation (ping-pong xA/xB)
    const int n4blocks   = (int)(((size_t)N_NODES * NCLS / 4 + 255) / 256);
    const int edgeblocks = (N_EDGES + 255) / 256;
    const float* src = h;
    for (int k = 0; k < KPROP; ++k) {
        float* dst = (k & 1) ? xB : xA;
        appnp_init_kernel<<<n4blocks, 256, 0, stream>>>(dst, h);
        appnp_edge_kernel<<<edgeblocks, 256, 0, stream>>>(rowp, colp, ev, src, dst);
        src = dst;
    }

    appnp_logsoftmax_kernel<<<(N_NODES + 255) / 256, 256, 0, stream>>>(src, out);
}